// PositionalTracebackDecoder_12455405158667
// MI455X (gfx1250) — compile-verified
//
#include <hip/hip_runtime.h>

typedef float v2f __attribute__((ext_vector_type(2)));
typedef float v8f __attribute__((ext_vector_type(8)));

#define VOCAB  256
#define DMODEL 128
#define TLEN   512
#define BATCH  8

__global__ __launch_bounds__(256)
void ptd_conv_wmma_kernel(const float* __restrict__ x,
                          const float* __restrict__ W,
                          const float* __restrict__ bias,
                          float* __restrict__ out) {
    const int lane = threadIdx.x & 31;
    const int wid  = blockIdx.x * 8 + (threadIdx.x >> 5);
    // wave tile decomposition: v fastest (store locality), then t_tile, then b
    const int v      = wid & (VOCAB - 1);
    const int t_tile = (wid >> 8) & 31;
    const int b      = wid >> 13;
    const int t0     = t_tile * 16;
    const int lm     = lane & 15;
    const int half   = lane >> 4;

    const float* xb = x + (size_t)b * TLEN * VOCAB;

    // ---- A fragment: 16x4 F32 (rows = 16 t values; K = {t-1, t, t+1, bias-one})
    // ISA layout: lanes 0-15 hold K=0 (v0) / K=1 (v1); lanes 16-31 hold K=2 / K=3.
    v2f a;
    const int tm = t0 + lm;
    if (half == 0) {
        a.x = (tm >= 1) ? xb[(tm - 1) * VOCAB + v] : 0.0f;       // tap t-1
        a.y = xb[tm * VOCAB + v];                                // tap t
    } else {
        a.x = (tm + 1 < TLEN) ? xb[(tm + 1) * VOCAB + v] : 0.0f; // tap t+1
        a.y = 1.0f;                                              // K=3 -> bias row
    }

    // mask source (center tap) for the 8 output rows this lane half owns;
    // uniform address per half-wave -> broadcast loads, hit WGP$/L2.
    float xc[8];
    #pragma unroll
    for (int j = 0; j < 8; ++j)
        xc[j] = xb[(t0 + j + half * 8) * VOCAB + v];

    const float* W0 = W + (0 * VOCAB + v) * DMODEL;
    const float* W1 = W + (1 * VOCAB + v) * DMODEL;
    const float* W2 = W + (2 * VOCAB + v) * DMODEL;
    const float* Bv = bias + v * DMODEL;

    // ---- Phase 1: all 8 WMMAs (conv + bias fused), plus per-d-tile posenc.
    v8f   c[DMODEL / 16];
    float pe[DMODEL / 16];
    #pragma unroll
    for (int dt = 0; dt < DMODEL / 16; ++dt) {
        const int d = dt * 16 + lm;

        // B fragment: 4x16 F32 (K x d-tile); K=3 row carries the bias
        v2f bb;
        if (half == 0) { bb.x = W0[d]; bb.y = W1[d]; }
        else           { bb.x = W2[d]; bb.y = Bv[d]; }

        v8f z = {};
        c[dt] = __builtin_amdgcn_wmma_f32_16x16x4_f32(
                    false, a, false, bb, (short)0, z, false, false);

        // positional encoding: d is fixed per lane -> one sin/cos per d-tile
        const int   j64  = (d < 64) ? d : d - 64;
        const float rate = __expf(-0.14391156642700418f * (float)j64); // 10000^(-j/64)
        const float ang  = (float)v * rate;
        pe[dt] = (d < 64) ? __sinf(ang) : __cosf(ang);
    }

    const size_t obase = (((size_t)(b * TLEN + t0) * VOCAB) + v) * DMODEL;
    const size_t trow  = (size_t)VOCAB * DMODEL;   // stride between t rows

    // ---- Phase 2: j outer (one mask compare per t-row), dt inner
    // (8 back-to-back stores at +64B offsets -> contiguous 512B per half-wave).
    #pragma unroll
    for (int j = 0; j < 8; ++j) {
        const bool m = (xc[j] != 0.0f);
        float* row = out + obase + (size_t)(j + half * 8) * trow + lm;
        #pragma unroll
        for (int dt = 0; dt < DMODEL / 16; ++dt) {
            float y = c[dt][j];
            // sigmoid: single v_exp_f32 + single v_rcp_f32 (no IEEE-div expansion)
            const float e = __builtin_amdgcn_exp2f(y * -1.4426950408889634f); // exp(-y)
            y = __builtin_amdgcn_rcpf(1.0f + e);
            y = __builtin_fmaf(y, 11.313708498984761f, pe[dt]);  // *sqrt(128)+PE
            y = m ? y : 0.0f;                                    // mask where x==0
            // 512MB stream > 192MB L2: non-temporal store
            __builtin_nontemporal_store(y, row + dt * 16);
        }
    }
}

extern "C" void kernel_launch(void* const* d_in, const int* in_sizes, int n_in,
                              void* d_out, int out_size, void* d_ws, size_t ws_size,
                              hipStream_t stream) {
    (void)in_sizes; (void)n_in; (void)out_size; (void)d_ws; (void)ws_size;
    const float* x  = (const float*)d_in[0];
    const float* W  = (const float*)d_in[1];
    const float* bi = (const float*)d_in[2];
    float* out      = (float*)d_out;

    // 8 (B) * 32 (t tiles) * 256 (v) = 65536 waves; 8 waves (256 thr) per block
    const int waves  = BATCH * (TLEN / 16) * VOCAB;
    const int blocks = waves / 8;
    ptd_conv_wmma_kernel<<<blocks, 256, 0, stream>>>(x, W, bi, out);
}